// TGN_74861279969393
// MI455X (gfx1250) — compile-verified
//
#include <hip/hip_runtime.h>
#include <hip/hip_bf16.h>

#define NUM_NODES 1000000
#define MEM_DIM   128
#define TIME_DIM  32
#define MSG_DIM   64
#define EDGES     16384
#define GRU_IN    352   // 2*MEM + MSG + TIME
#define H3        384   // 3*MEM

typedef __attribute__((ext_vector_type(2))) float v2f;
typedef __attribute__((ext_vector_type(8))) float v8f;

static __device__ __forceinline__ v8f wmma_f32(v2f a, v2f b, v8f c) {
    // V_WMMA_F32_16X16X4_F32 : D = A(16x4,f32) * B(4x16,f32) + C(16x16,f32)
    return __builtin_amdgcn_wmma_f32_16x16x4_f32(
        /*neg_a=*/false, a, /*neg_b=*/false, b,
        /*c_mod=*/(short)0, c, /*reuse_a=*/false, /*reuse_b=*/false);
}

static __device__ __forceinline__ float sigmoidf_(float x) {
    return 1.0f / (1.0f + __expf(-x));
}

// ---------------- K1: reset best_t / best_pos for touched nodes ----------------
__global__ void k_reset(const int* __restrict__ src, const int* __restrict__ dst,
                        const int* __restrict__ neg, int* __restrict__ best_t,
                        int* __restrict__ best_pos) {
    int i = blockIdx.x * blockDim.x + threadIdx.x;   // [0, 3E)
    if (i >= 3 * EDGES) return;
    int node;
    if (i < EDGES)            node = src[i];
    else if (i < 2 * EDGES)   node = dst[i - EDGES];
    else                      node = neg[i - 2 * EDGES];
    best_t[node]   = -1;     // all racers write the same value
    best_pos[node] = -1;
}

// ---------------- K2: best_t[node] = max t over its messages ----------------
__global__ void k_maxt(const int* __restrict__ src, const int* __restrict__ dst,
                       const int* __restrict__ t, int* __restrict__ best_t) {
    int i = blockIdx.x * blockDim.x + threadIdx.x;   // [0, 2E)
    if (i >= 2 * EDGES) return;
    int e    = (i < EDGES) ? i : i - EDGES;
    int node = (i < EDGES) ? src[e] : dst[e];
    atomicMax(&best_t[node], t[e]);
}

// ---------------- K3: best_pos = max row index among rows hitting best_t ------
__global__ void k_maxpos(const int* __restrict__ src, const int* __restrict__ dst,
                         const int* __restrict__ t, const int* __restrict__ best_t,
                         int* __restrict__ best_pos) {
    int i = blockIdx.x * blockDim.x + threadIdx.x;   // [0, 2E)
    if (i >= 2 * EDGES) return;
    int e    = (i < EDGES) ? i : i - EDGES;
    int node = (i < EDGES) ? src[e] : dst[e];
    if (t[e] == best_t[node]) atomicMax(&best_pos[node], i);
}

// ---------------- K4: fused message-build + dual GEMM (fp32 WMMA) + GRU gates -
// One workgroup (8 waves, 256 threads) per 16-row tile of the 2E message rows.
// LDS holds the 16x352 message tile; its first 128 columns are h_old.
__global__ void __launch_bounds__(256)
k_gru_wmma(const float* __restrict__ memory, const int* __restrict__ last_update,
           const int* __restrict__ src, const int* __restrict__ dst,
           const int* __restrict__ t, const float* __restrict__ raw_msg,
           const float* __restrict__ time_w, const float* __restrict__ time_b,
           const float* __restrict__ w_ih, const float* __restrict__ w_hh,
           const float* __restrict__ b_ih, const float* __restrict__ b_hh,
           float* __restrict__ Hnew) {
    __shared__ float Xl[16 * GRU_IN];                // 22528 B

    const int tid  = threadIdx.x;
    const int wave = tid >> 5;
    const int lane = tid & 31;
    const int half = lane >> 4;                      // 0: K lo pair, 1: K hi pair
    const int l    = lane & 15;
    const int row0 = blockIdx.x * 16;

    // ---- build the message tile in LDS (gathers + time encoding) ----
    for (int q = tid; q < 16 * GRU_IN; q += 256) {
        int i = q / GRU_IN, c = q - i * GRU_IN;
        int e = row0 + i;
        int edge, an, bn;
        if (e < EDGES) { edge = e;         an = src[edge]; bn = dst[edge]; }
        else           { edge = e - EDGES; an = dst[edge]; bn = src[edge]; }
        float v;
        if (c < MEM_DIM)            v = memory[(size_t)an * MEM_DIM + c];
        else if (c < 2 * MEM_DIM)   v = memory[(size_t)bn * MEM_DIM + (c - MEM_DIM)];
        else if (c < 2 * MEM_DIM + MSG_DIM)
                                    v = raw_msg[(size_t)edge * MSG_DIM + (c - 2 * MEM_DIM)];
        else {
            int j = c - (2 * MEM_DIM + MSG_DIM);
            float dt = (float)(t[edge] - last_update[an]);
            v = cosf(dt * time_w[j] + time_b[j]);
        }
        Xl[i * GRU_IN + c] = v;
    }
    __syncthreads();

    const int j0 = wave * 16;                        // this wave's gate columns
    v8f acc[6] = {};                                 // gi_r,gi_z,gi_n, gh_r,gh_z,gh_n

    // ---- gi = msg @ w_ih.T : K = 352 ----
    for (int ks = 0; ks < GRU_IN; ks += 4) {
        const int kk = ks + 2 * half;
        v2f a; a.x = Xl[l * GRU_IN + kk]; a.y = Xl[l * GRU_IN + kk + 1];
#pragma unroll
        for (int g = 0; g < 3; ++g) {
            const float* wp = w_ih + (size_t)(g * MEM_DIM + j0 + l) * GRU_IN + kk;
            v2f b; b.x = wp[0]; b.y = wp[1];
            acc[g] = wmma_f32(a, b, acc[g]);
        }
    }
    // ---- gh = h_old @ w_hh.T : K = 128 (h_old = first 128 cols of Xl) ----
    for (int ks = 0; ks < MEM_DIM; ks += 4) {
        const int kk = ks + 2 * half;
        v2f a; a.x = Xl[l * GRU_IN + kk]; a.y = Xl[l * GRU_IN + kk + 1];
#pragma unroll
        for (int g = 0; g < 3; ++g) {
            const float* wp = w_hh + (size_t)(g * MEM_DIM + j0 + l) * MEM_DIM + kk;
            v2f b; b.x = wp[0]; b.y = wp[1];
            acc[3 + g] = wmma_f32(a, b, acc[3 + g]);
        }
    }

    // ---- gates + h_new (C/D layout: vgpr r, lane (half,l) = element (r+8*half, j0+l)) ----
    const float bi_r = b_ih[j0 + l],           bh_r = b_hh[j0 + l];
    const float bi_z = b_ih[MEM_DIM + j0 + l], bh_z = b_hh[MEM_DIM + j0 + l];
    const float bi_n = b_ih[2*MEM_DIM + j0 + l], bh_n = b_hh[2*MEM_DIM + j0 + l];
#pragma unroll
    for (int r = 0; r < 8; ++r) {
        const int m = r + 8 * half;
        float rg = sigmoidf_((acc[0][r] + bi_r) + (acc[3][r] + bh_r));
        float zg = sigmoidf_((acc[1][r] + bi_z) + (acc[4][r] + bh_z));
        float ng = tanhf((acc[2][r] + bi_n) + rg * (acc[5][r] + bh_n));
        float ho = Xl[m * GRU_IN + j0 + l];
        Hnew[(size_t)(row0 + m) * MEM_DIM + j0 + l] = (1.0f - zg) * ng + zg * ho;
    }
}

// ---------------- K5: link prediction (MLP) with best_pos indirection --------
__global__ void __launch_bounds__(128)
k_linkpred(const float* __restrict__ memory, const int* __restrict__ src,
           const int* __restrict__ pos_dst, const int* __restrict__ neg_dst,
           const int* __restrict__ best_pos, const float* __restrict__ Hnew,
           const float* __restrict__ w1, const float* __restrict__ b1,
           const float* __restrict__ w2, const float* __restrict__ b2,
           float* __restrict__ out) {
    __shared__ float cat[2 * MEM_DIM];
    __shared__ float red[MEM_DIM];
    const int bid  = blockIdx.x;          // [0, 2E): first E = pos, next E = neg
    const int tid  = threadIdx.x;         // 128
    const int pred = bid >> 14;           // bid / EDGES
    const int e    = bid & (EDGES - 1);

    const int an = src[e];
    const int bn = pred ? neg_dst[e] : pos_dst[e];
    const int bpa = best_pos[an], bpb = best_pos[bn];
    const float* ea = (bpa >= 0) ? (Hnew + (size_t)bpa * MEM_DIM)
                                 : (memory + (size_t)an * MEM_DIM);
    const float* eb = (bpb >= 0) ? (Hnew + (size_t)bpb * MEM_DIM)
                                 : (memory + (size_t)bn * MEM_DIM);
    cat[tid]           = ea[tid];
    cat[MEM_DIM + tid] = eb[tid];
    __syncthreads();

    const float* wr = w1 + (size_t)tid * (2 * MEM_DIM);
    float s = b1[tid];
#pragma unroll 8
    for (int k = 0; k < 2 * MEM_DIM; ++k) s += cat[k] * wr[k];
    red[tid] = fmaxf(s, 0.0f) * w2[tid];
    __syncthreads();
#pragma unroll
    for (int off = 64; off > 0; off >>= 1) {
        if (tid < off) red[tid] += red[tid + off];
        __syncthreads();
    }
    if (tid == 0) out[bid] = red[0] + b2[0];
}

extern "C" void kernel_launch(void* const* d_in, const int* in_sizes, int n_in,
                              void* d_out, int out_size, void* d_ws, size_t ws_size,
                              hipStream_t stream) {
    const float* memory   = (const float*)d_in[0];
    const int*   last_upd = (const int*)  d_in[1];
    const int*   src      = (const int*)  d_in[2];
    const int*   pos_dst  = (const int*)  d_in[3];
    const int*   neg_dst  = (const int*)  d_in[4];
    const int*   t        = (const int*)  d_in[5];
    const float* raw_msg  = (const float*)d_in[6];
    const float* time_w   = (const float*)d_in[7];
    const float* time_b   = (const float*)d_in[8];
    const float* w_ih     = (const float*)d_in[9];
    const float* w_hh     = (const float*)d_in[10];
    const float* b_ih     = (const float*)d_in[11];
    const float* b_hh     = (const float*)d_in[12];
    const float* lp_w1    = (const float*)d_in[13];
    const float* lp_b1    = (const float*)d_in[14];
    const float* lp_w2    = (const float*)d_in[15];
    const float* lp_b2    = (const float*)d_in[16];
    float* out = (float*)d_out;

    // workspace: best_t [1M i32] | best_pos [1M i32] | Hnew [2E x 128 f32]  (~24.8 MB)
    char* ws = (char*)d_ws;
    int*   best_t   = (int*)(ws);
    int*   best_pos = (int*)(ws + 4000000);
    float* Hnew     = (float*)(ws + 8000000);

    k_reset <<<(3 * EDGES) / 256, 256, 0, stream>>>(src, pos_dst, neg_dst, best_t, best_pos);
    k_maxt  <<<(2 * EDGES) / 256, 256, 0, stream>>>(src, pos_dst, t, best_t);
    k_maxpos<<<(2 * EDGES) / 256, 256, 0, stream>>>(src, pos_dst, t, best_t, best_pos);
    k_gru_wmma<<<(2 * EDGES) / 16, 256, 0, stream>>>(memory, last_upd, src, pos_dst, t,
                                                     raw_msg, time_w, time_b,
                                                     w_ih, w_hh, b_ih, b_hh, Hnew);
    k_linkpred<<<2 * EDGES, 128, 0, stream>>>(memory, src, pos_dst, neg_dst,
                                              best_pos, Hnew,
                                              lp_w1, lp_b1, lp_w2, lp_b2, out);
}